// BilinearInterpolation_38706245271954
// MI455X (gfx1250) — compile-verified
//
#include <hip/hip_runtime.h>

// Bilinear spatial-transformer sampling, MI455X (gfx1250).
// B=32, H=W=256, C=32, fp32. Memory-bound gather: optimized for
// 128B global_load_b128 gathers + nontemporal (TH=NT) b128 stores.

typedef float v4f __attribute__((ext_vector_type(4)));

#define B_  32
#define H_  256
#define W_  256
#define C_  32
#define CG_ (C_ / 4)              // 8 channel-groups of float4 per pixel

__global__ __launch_bounds__(256) void
BilinearInterpolation_38706245271954_kernel(const float* __restrict__ image,
                                            const float* __restrict__ theta,
                                            float* __restrict__ out)
{
    const int tid = blockIdx.x * 256 + threadIdx.x;
    const int cg  = tid & (CG_ - 1);     // which float4 of the 32 channels
    const int pix = tid >> 3;            // flat output pixel (b,h,w)
    const int w   = pix & (W_ - 1);
    const int h   = (pix >> 8) & (H_ - 1);
    // 32 pixels per block, 65536 pixels per batch image -> b is uniform
    // across the block; force the scalar path so theta comes in via SMEM.
    const int b   = __builtin_amdgcn_readfirstlane(pix >> 16);

    const float* __restrict__ t = theta + b * 6;
    const float t0 = t[0], t1 = t[1], t2 = t[2];
    const float t3 = t[3], t4 = t[4], t5 = t[5];

    // regular grid in [-1,1]: linspace(-1,1,256) -> step 2/255
    const float step = 2.0f / 255.0f;
    const float xg = fmaf((float)w, step, -1.0f);
    const float yg = fmaf((float)h, step, -1.0f);

    // affine transform (einsum bij,jn->bin)
    const float xs = fmaf(t0, xg, fmaf(t1, yg, t2));
    const float ys = fmaf(t3, xg, fmaf(t4, yg, t5));

    // [-1,1] -> pixel coords, scale by W (not W-1), per reference
    const float x = (xs + 1.0f) * (0.5f * (float)W_);
    const float y = (ys + 1.0f) * (0.5f * (float)H_);

    // truncation toward zero (matches astype(int32)); x1 = x0+1 BEFORE clip
    int x0 = (int)x;
    int y0 = (int)y;
    int x1 = x0 + 1;
    int y1 = y0 + 1;
    x0 = min(max(x0, 0), W_ - 1);
    x1 = min(max(x1, 0), W_ - 1);
    y0 = min(max(y0, 0), H_ - 1);
    y1 = min(max(y1, 0), H_ - 1);

    // bilinear weights from CLIPPED corner coords (matches reference)
    const float x0f = (float)x0, x1f = (float)x1;
    const float y0f = (float)y0, y1f = (float)y1;
    const float wa = (x1f - x) * (y1f - y);
    const float wb = (x1f - x) * (y - y0f);
    const float wc = (x - x0f) * (y1f - y);
    const float wd = (x - x0f) * (y - y0f);

    // gather the 4 corner pixels: 8 lanes x float4 = 128B per corner pixel
    const size_t batchBase = (size_t)b * (H_ * W_ * C_) + (size_t)(cg * 4);
    const float* __restrict__ imgB = image + batchBase;

    const v4f pa = *(const v4f*)(imgB + ((size_t)y0 * W_ + x0) * C_);
    const v4f pb = *(const v4f*)(imgB + ((size_t)y1 * W_ + x0) * C_);
    const v4f pc = *(const v4f*)(imgB + ((size_t)y0 * W_ + x1) * C_);
    const v4f pd = *(const v4f*)(imgB + ((size_t)y1 * W_ + x1) * C_);

    v4f r;
    r.x = fmaf(wa, pa.x, fmaf(wb, pb.x, fmaf(wc, pc.x, wd * pd.x)));
    r.y = fmaf(wa, pa.y, fmaf(wb, pb.y, fmaf(wc, pc.y, wd * pd.y)));
    r.z = fmaf(wa, pa.z, fmaf(wb, pb.z, fmaf(wc, pc.z, wd * pd.z)));
    r.w = fmaf(wa, pa.w, fmaf(wb, pb.w, fmaf(wc, pc.w, wd * pd.w)));

    // output is write-once streaming (256 MB): nontemporal store keeps the
    // 192MB L2 free for the gather working set (8MB per batch image)
    float* __restrict__ o =
        out + (size_t)b * (H_ * W_ * C_) + ((size_t)h * W_ + w) * C_ + cg * 4;
    __builtin_nontemporal_store(r, (v4f*)o);
}

extern "C" void kernel_launch(void* const* d_in, const int* in_sizes, int n_in,
                              void* d_out, int out_size, void* d_ws, size_t ws_size,
                              hipStream_t stream)
{
    const float* image = (const float*)d_in[0];   // [32,256,256,32] fp32
    const float* theta = (const float*)d_in[1];   // [32,6] fp32
    float* out = (float*)d_out;                   // [32,256,256,32] fp32

    // total threads = B*H*W*C/4 = 16,777,216 -> 65536 blocks of 256
    const int totalThreads = B_ * H_ * W_ * CG_;
    const int block = 256;
    const int grid  = totalThreads / block;       // 65536

    BilinearInterpolation_38706245271954_kernel<<<grid, block, 0, stream>>>(
        image, theta, out);
}